// GCN3_91036126806358
// MI455X (gfx1250) — compile-verified
//
#include <hip/hip_runtime.h>
#include <hip/hip_bf16.h>
#include <math.h>

typedef float v2f __attribute__((ext_vector_type(2)));
typedef float v8f __attribute__((ext_vector_type(8)));

#define N_NODES 20000
#define N_EDGES 320000
#define NFEAT   512
#define NHID    256
#define NCLASS  64

__device__ __forceinline__ float selu_f(float x) {
    const float scale = 1.0507009873554805f;
    const float alpha = 1.6732632423543772f;
    return x > 0.0f ? scale * x : scale * alpha * (expf(x) - 1.0f);
}

// ---------------- zero fill (guarded) ----------------
__global__ void zero_i32(int* __restrict__ p, int n) {
    int i = blockIdx.x * blockDim.x + threadIdx.x;
    if (i < n) p[i] = 0;
}

// ---------------- CSR build: per-row edge counts ----------------
__global__ void count_edges(const int* __restrict__ ei, int* __restrict__ cnt) {
    int e = blockIdx.x * blockDim.x + threadIdx.x;   // exact grid: N_EDGES
    atomicAdd(&cnt[ei[e]], 1);
}

// ---------------- CSR build: exclusive scan of counts (1 block, 256 thr) ----
__global__ void build_start(const int* __restrict__ cnt,
                            int* __restrict__ start, int* __restrict__ cursor) {
    __shared__ int part[256];
    __shared__ int base[256];
    const int CH = (N_NODES + 255) / 256;   // 79
    int t = threadIdx.x;
    int lo = t * CH;
    int hi = lo + CH < N_NODES ? lo + CH : N_NODES;
    int s = 0;
    for (int i = lo; i < hi; ++i) s += cnt[i];
    part[t] = s;
    __syncthreads();
    if (t == 0) {
        int acc = 0;
        for (int i = 0; i < 256; ++i) { base[i] = acc; acc += part[i]; }
    }
    __syncthreads();
    int acc = base[t];
    for (int i = lo; i < hi; ++i) {
        start[i]  = acc;
        cursor[i] = acc;
        acc += cnt[i];
    }
}

// ---------------- CSR build: scatter edge ids into row order ----------------
__global__ void fill_order(const int* __restrict__ ei,
                           int* __restrict__ cursor, int* __restrict__ order) {
    int e = blockIdx.x * blockDim.x + threadIdx.x;   // exact grid: N_EDGES
    int r = ei[e];
    int p = atomicAdd(&cursor[r], 1);
    order[p] = e;
}

// ---------------- fp32 WMMA GEMM: C[M,N] = A[M,K] @ B[K,N] ----------------
// One wave per 16x16 tile; exact grids so EXEC is all-1s (WMMA requirement).
__global__ void gemm_wmma_f32(const float* __restrict__ A,
                              const float* __restrict__ B,
                              float* __restrict__ C,
                              int N, int K) {
    const int wave = blockIdx.x * (blockDim.x >> 5) + (threadIdx.x >> 5);
    const int lane = threadIdx.x & 31;
    const int ntiles = N >> 4;
    const int tm = wave / ntiles;
    const int tn = wave - tm * ntiles;

    const int half = lane >> 4;
    const int r    = lane & 15;

    const float* arow = A + (size_t)(tm * 16 + r) * K + 2 * half;
    const float* bcol = B + (size_t)(2 * half) * N + tn * 16 + r;

    v8f c = {0.f, 0.f, 0.f, 0.f, 0.f, 0.f, 0.f, 0.f};

#pragma unroll 4
    for (int k = 0; k < K; k += 4) {
        v2f a = *(const v2f*)(arow + k);
        v2f b;
        b.x = bcol[(size_t)k * N];
        b.y = bcol[(size_t)(k + 1) * N];
        c = __builtin_amdgcn_wmma_f32_16x16x4_f32(
                false, a, false, b, (short)0, c, false, false);
    }

    float* crow = C + (size_t)(tm * 16 + half * 8) * N + tn * 16 + r;
#pragma unroll
    for (int i = 0; i < 8; ++i)
        crow[(size_t)i * N] = c[i];
}

// ---------------- per-node aggregation, d=256: h = selu(sum w*sup[col] + b) --
// One wave per node. Lane L covers features [4L,4L+3] and [4L+128,4L+131].
__global__ void aggregate256(const int* __restrict__ start, const int* __restrict__ cnt,
                             const int* __restrict__ order,
                             const int* __restrict__ ei, const float* __restrict__ ew,
                             const float* __restrict__ sup,
                             const float* __restrict__ bias,
                             float* __restrict__ h) {
    int node = blockIdx.x * (blockDim.x >> 5) + (threadIdx.x >> 5);  // exact: N_NODES waves
    int lane = threadIdx.x & 31;
    int c0 = lane * 4;

    int s0 = start[node];
    int n  = cnt[node];

    float4 acc0 = {0.f, 0.f, 0.f, 0.f};
    float4 acc1 = {0.f, 0.f, 0.f, 0.f};

    for (int q = 0; q < n; ++q) {
        int   e   = order[s0 + q];
        int   col = ei[N_EDGES + e];
        float w   = ew[e];
        const float* src = sup + ((size_t)col << 8) + c0;
        if (q + 1 < n) {   // prefetch next source row (global_prefetch_b8)
            int e2 = order[s0 + q + 1];
            __builtin_prefetch(sup + ((size_t)ei[N_EDGES + e2] << 8) + c0, 0, 1);
        }
        float4 a = *(const float4*)(src);
        float4 b = *(const float4*)(src + 128);
        acc0.x = fmaf(w, a.x, acc0.x); acc0.y = fmaf(w, a.y, acc0.y);
        acc0.z = fmaf(w, a.z, acc0.z); acc0.w = fmaf(w, a.w, acc0.w);
        acc1.x = fmaf(w, b.x, acc1.x); acc1.y = fmaf(w, b.y, acc1.y);
        acc1.z = fmaf(w, b.z, acc1.z); acc1.w = fmaf(w, b.w, acc1.w);
    }

    float4 bb0 = *(const float4*)(bias + c0);
    float4 bb1 = *(const float4*)(bias + c0 + 128);
    float4 o0, o1;
    o0.x = selu_f(acc0.x + bb0.x); o0.y = selu_f(acc0.y + bb0.y);
    o0.z = selu_f(acc0.z + bb0.z); o0.w = selu_f(acc0.w + bb0.w);
    o1.x = selu_f(acc1.x + bb1.x); o1.y = selu_f(acc1.y + bb1.y);
    o1.z = selu_f(acc1.z + bb1.z); o1.w = selu_f(acc1.w + bb1.w);

    float* dst = h + ((size_t)node << 8) + c0;
    *(float4*)(dst)       = o0;
    *(float4*)(dst + 128) = o1;
}

// ---------------- per-node aggregation, d=64 ----------------
// One wave per node. Lane L covers features [2L, 2L+1].
__global__ void aggregate64(const int* __restrict__ start, const int* __restrict__ cnt,
                            const int* __restrict__ order,
                            const int* __restrict__ ei, const float* __restrict__ ew,
                            const float* __restrict__ sup,
                            const float* __restrict__ bias,
                            float* __restrict__ h) {
    int node = blockIdx.x * (blockDim.x >> 5) + (threadIdx.x >> 5);
    int lane = threadIdx.x & 31;
    int c0 = lane * 2;

    int s0 = start[node];
    int n  = cnt[node];

    float2 acc = {0.f, 0.f};
    for (int q = 0; q < n; ++q) {
        int   e   = order[s0 + q];
        int   col = ei[N_EDGES + e];
        float w   = ew[e];
        float2 a = *(const float2*)(sup + ((size_t)col << 6) + c0);
        acc.x = fmaf(w, a.x, acc.x);
        acc.y = fmaf(w, a.y, acc.y);
    }

    float2 bb = *(const float2*)(bias + c0);
    float2 o;
    o.x = selu_f(acc.x + bb.x);
    o.y = selu_f(acc.y + bb.y);
    *(float2*)(h + ((size_t)node << 6) + c0) = o;
}

// ---------------- column mean over nodes ----------------
__global__ void mean_cols(const float* __restrict__ h, float* __restrict__ pooled) {
    __shared__ float sm[256];
    int c = blockIdx.x, t = threadIdx.x;
    float s = 0.0f;
    for (int r = t; r < N_NODES; r += 256)
        s += h[(size_t)r * NCLASS + c];
    sm[t] = s; __syncthreads();
    for (int o = 128; o > 0; o >>= 1) {
        if (t < o) sm[t] += sm[t + o];
        __syncthreads();
    }
    if (t == 0) pooled[c] = sm[0] * (1.0f / (float)N_NODES);
}

// ---------------- selu + log_softmax, 64 threads ----------------
__global__ void finalize(const float* __restrict__ pooled, float* __restrict__ out) {
    __shared__ float s[NCLASS];
    int t = threadIdx.x;
    float v = selu_f(pooled[t]);
    s[t] = v; __syncthreads();
    float m = -INFINITY;
    for (int i = 0; i < NCLASS; ++i) m = fmaxf(m, s[i]);
    float sum = 0.0f;
    for (int i = 0; i < NCLASS; ++i) sum += expf(s[i] - m);
    out[t] = v - m - logf(sum);
}

extern "C" void kernel_launch(void* const* d_in, const int* in_sizes, int n_in,
                              void* d_out, int out_size, void* d_ws, size_t ws_size,
                              hipStream_t stream) {
    const float* x  = (const float*)d_in[0];   // [20000, 512]
    const int*   ei = (const int*)  d_in[1];   // [2, 320000]
    const float* ew = (const float*)d_in[2];   // [320000]
    const float* W1 = (const float*)d_in[3];   // [512, 256]
    const float* b1 = (const float*)d_in[4];   // [256]
    const float* W2 = (const float*)d_in[5];   // [256, 64]
    const float* b2 = (const float*)d_in[6];   // [64]
    float* out = (float*)d_out;                // [64]

    float* sup1   = (float*)d_ws;                        // 20000*256
    float* h1     = sup1 + (size_t)N_NODES * NHID;       // 20000*256
    float* sup2   = h1   + (size_t)N_NODES * NHID;       // 20000*64
    float* h2     = sup2 + (size_t)N_NODES * NCLASS;     // 20000*64
    float* pooled = h2   + (size_t)N_NODES * NCLASS;     // 64
    int*   cnt    = (int*)(pooled + NCLASS);             // 20000
    int*   startp = cnt    + N_NODES;                    // 20000
    int*   cursor = startp + N_NODES;                    // 20000
    int*   order  = cursor + N_NODES;                    // 320000

    const int BT = 256;

    // ---- build CSR of destination rows (shared by both layers) ----
    zero_i32<<<(N_NODES + BT - 1) / BT, BT, 0, stream>>>(cnt, N_NODES);
    count_edges<<<N_EDGES / BT, BT, 0, stream>>>(ei, cnt);
    build_start<<<1, BT, 0, stream>>>(cnt, startp, cursor);
    fill_order<<<N_EDGES / BT, BT, 0, stream>>>(ei, cursor, order);

    // ---- layer 1: support1 = x @ W1 (WMMA), then gather-aggregate + bias + selu
    gemm_wmma_f32<<<(N_NODES / 16) * (NHID / 16) / 8, BT, 0, stream>>>(x, W1, sup1, NHID, NFEAT);
    aggregate256<<<N_NODES / 8, BT, 0, stream>>>(startp, cnt, order, ei, ew, sup1, b1, h1);

    // ---- layer 2: support2 = h1 @ W2 (WMMA), then gather-aggregate + bias + selu
    gemm_wmma_f32<<<(N_NODES / 16) * (NCLASS / 16) / 8, BT, 0, stream>>>(h1, W2, sup2, NCLASS, NHID);
    aggregate64<<<N_NODES / 8, BT, 0, stream>>>(startp, cnt, order, ei, ew, sup2, b2, h2);

    // ---- pooled mean -> selu -> log_softmax ----
    mean_cols<<<NCLASS, BT, 0, stream>>>(h2, pooled);
    finalize<<<1, NCLASS, 0, stream>>>(pooled, out);
}